// Criterion4OL_38732015075811
// MI455X (gfx1250) — compile-verified
//
#include <hip/hip_runtime.h>
#include <hip/hip_bf16.h>
#include <float.h>

#define NSTAGE 3
#define NBATCH 32
#define NPRI   2000
#define NLANE  4
#define NPTS   72
#define DD     78
#define TILE   32          // priors per async tile (even -> tile bytes multiple of 16)
#define NSB    (NSTAGE * NBATCH)   // 96

typedef float v2f __attribute__((ext_vector_type(2)));
typedef float v8f __attribute__((ext_vector_type(8)));

// ---------------------------------------------------------------------------
// Kernel 1: per (branch, s, b) cost matrix + greedy assignment.
// Priors streamed global->LDS with gfx1250 async-to-LDS (ASYNCcnt), double
// buffered; full cost[2000][4] stays in LDS for the 4 argmin rounds.
// ---------------------------------------------------------------------------
__global__ __launch_bounds__(256) void assign_kernel(
    const float* __restrict__ predF, const float* __restrict__ predS,
    const float* __restrict__ gt, int* __restrict__ rowsWS)
{
    __shared__ float cost[NPRI][NLANE];                              // 32000 B
    __shared__ __attribute__((aligned(16))) float tileBuf[2][TILE * DD]; // 19968 B
    __shared__ float tScaled[NLANE][NPTS];
    __shared__ float tGeo[NLANE][4];
    __shared__ float rVal[256];
    __shared__ int   rIdx[256];
    __shared__ int   rowsSh[NLANE];

    const int id     = blockIdx.x;          // 0..191
    const int branch = id / NSB;
    const int sb     = id % NSB;            // = s*B + b (matches reshape order)
    const int b      = sb % NBATCH;
    const float* preds = (branch ? predS : predF) + (size_t)sb * NPRI * DD;
    const float* g     = gt + (size_t)b * NLANE * DD;
    const int tid = threadIdx.x;

    // GT-derived tables
    for (int i = tid; i < NLANE * NPTS; i += 256) {
        int l = i / NPTS, k = i % NPTS;
        tScaled[l][k] = g[l * DD + 6 + k] * (1.0f / 799.0f);
    }
    if (tid < NLANE * 4) {
        int l = tid >> 2, j = tid & 3;
        tGeo[l][j] = g[l * DD + 2 + j];
    }
    __syncthreads();

    // --- async tile issue: contiguous rows*312B region as b128 chunks -------
    auto issueTile = [&](int t, int parity) {
        int rows   = NPRI - t * TILE; if (rows > TILE) rows = TILE;
        int chunks = rows * DD * 4 / 16;               // rows even -> exact
        const float* base = preds + (size_t)t * TILE * DD;  // 16B-aligned offset
        unsigned ldsBase = (unsigned)(unsigned long long)(&tileBuf[parity][0]);
        for (int it = 0; it < 3; ++it) {               // 3 issues per wave, always
            int c  = tid + it * 256;
            int cc = (c < chunks) ? c : 0;             // clamp: benign dup of chunk 0
            unsigned voff  = (unsigned)(cc * 16);
            unsigned laddr = ldsBase + voff;
            asm volatile("global_load_async_to_lds_b128 %0, %1, %2"
                         :: "v"(laddr), "v"(voff), "s"(base) : "memory");
        }
    };

    issueTile(0, 0);
    const int T = (NPRI + TILE - 1) / TILE;            // 63
    for (int t = 0; t < T; ++t) {
        int par = t & 1;
        if (t + 1 < T) {
            issueTile(t + 1, par ^ 1);
            asm volatile("s_wait_asynccnt 3" ::: "memory");   // tile t's 3 done
        } else {
            asm volatile("s_wait_asynccnt 0" ::: "memory");
        }
        __syncthreads();

        // 8 lanes per prior: chunk = tid&7 covers 9 offsets
        int slot  = tid >> 3;
        int chunk = tid & 7;
        int n     = t * TILE + slot;
        int rowsInTile = NPRI - t * TILE; if (rowsInTile > TILE) rowsInTile = TILE;

        float off0 = 0.f, off1 = 0.f, off2 = 0.f, off3 = 0.f;
        if (slot < rowsInTile) {
            const float* pr = &tileBuf[par][slot * DD];
            float pk[9];
            #pragma unroll
            for (int k = 0; k < 9; ++k) pk[k] = pr[6 + chunk * 9 + k];
            #pragma unroll
            for (int k = 0; k < 9; ++k) {
                int kk = chunk * 9 + k;
                off0 += fabsf(pk[k] - tScaled[0][kk]);
                off1 += fabsf(pk[k] - tScaled[1][kk]);
                off2 += fabsf(pk[k] - tScaled[2][kk]);
                off3 += fabsf(pk[k] - tScaled[3][kk]);
            }
        }
        #pragma unroll
        for (int m = 1; m < 8; m <<= 1) {              // reduce within 8-lane group
            off0 += __shfl_xor(off0, m);
            off1 += __shfl_xor(off1, m);
            off2 += __shfl_xor(off2, m);
            off3 += __shfl_xor(off3, m);
        }
        if (chunk == 0 && slot < rowsInTile) {
            const float* pr = &tileBuf[par][slot * DD];
            float p0 = pr[0], p1 = pr[1];
            float mx = fmaxf(p0, p1);
            float e0 = __expf(p0 - mx), e1 = __expf(p1 - mx);
            float score = e1 / (e0 + e1);
            float pg0 = pr[2], pg1 = pr[3], pg2 = pr[4], pg3 = pr[5];
            float offs[4] = {off0, off1, off2, off3};
            #pragma unroll
            for (int l = 0; l < NLANE; ++l) {
                float geo = fabsf(pg0 - tGeo[l][0]) + fabsf(pg1 - tGeo[l][1]) +
                            fabsf(pg2 - tGeo[l][2]) + fabsf(pg3 - tGeo[l][3]);
                cost[n][l] = geo + offs[l] * (1.0f / 72.0f) - score;
            }
        }
        __syncthreads();
    }

    // --- greedy argmin per GT column, first-occurrence tie-break ------------
    for (int l = 0; l < NLANE; ++l) {
        float best = FLT_MAX; int bidx = NPRI;
        for (int n = tid; n < NPRI; n += 256) {
            bool used = false;
            for (int j = 0; j < l; ++j) used |= (rowsSh[j] == n);
            float v = used ? FLT_MAX : cost[n][l];
            if (v < best) { best = v; bidx = n; }      // ascending scan: first wins
        }
        rVal[tid] = best; rIdx[tid] = bidx;
        __syncthreads();
        for (int o = 128; o > 0; o >>= 1) {
            if (tid < o) {
                float v2 = rVal[tid + o]; int i2 = rIdx[tid + o];
                if (v2 < rVal[tid] || (v2 == rVal[tid] && i2 < rIdx[tid])) {
                    rVal[tid] = v2; rIdx[tid] = i2;
                }
            }
            __syncthreads();
        }
        if (tid == 0) rowsSh[l] = rIdx[0];
        __syncthreads();
    }
    if (tid < NLANE) rowsWS[(branch * NSB + sb) * NLANE + tid] = rowsSh[tid];
}

// ---------------------------------------------------------------------------
// Kernel 2: focal classification loss per prior (sum over the 96 (s,b) pairs)
// ---------------------------------------------------------------------------
__global__ __launch_bounds__(256) void cls_kernel(
    const float* __restrict__ predF, const float* __restrict__ predS,
    const int* __restrict__ rowsWS, float* __restrict__ lossWS)
{
    const int branch = blockIdx.x >> 3;       // 8 blocks of 256 cover 2000 priors
    const int nb     = blockIdx.x & 7;
    const float* preds = branch ? predS : predF;
    __shared__ int rsh[NSB * NLANE];
    for (int i = threadIdx.x; i < NSB * NLANE; i += 256)
        rsh[i] = rowsWS[branch * NSB * NLANE + i];
    __syncthreads();

    int n = nb * 256 + threadIdx.x;
    if (n >= NPRI) return;
    float acc = 0.f;
    for (int sb = 0; sb < NSB; ++sb) {
        const float* p = preds + ((size_t)sb * NPRI + n) * DD;
        float p0 = p[0], p1 = p[1];
        bool mt = (rsh[sb*4] == n) | (rsh[sb*4+1] == n) |
                  (rsh[sb*4+2] == n) | (rsh[sb*4+3] == n);
        float mx  = fmaxf(p0, p1);
        float lse = mx + __logf(__expf(p0 - mx) + __expf(p1 - mx));
        float logpt = (mt ? p1 : p0) - lse;
        float pt    = __expf(logpt);
        float alpha = mt ? 0.9f : 0.1f;
        float om    = 1.0f - pt;
        acc += -alpha * om * om * logpt;
    }
    lossWS[branch * NPRI + n] = 2.0f * acc * (1.0f / 96.0f);   // CLS_W / (B*S)
}

// ---------------------------------------------------------------------------
// Kernel 3: smooth-L1 + line-IoU over matched priors, added at rows_last
// ---------------------------------------------------------------------------
__global__ __launch_bounds__(512) void regiou_kernel(
    const float* __restrict__ predF, const float* __restrict__ predS,
    const float* __restrict__ gt, const int* __restrict__ rowsWS,
    float* __restrict__ lossWS)
{
    const int branch = blockIdx.x;
    const float* preds = branch ? predS : predF;
    __shared__ float regP[512], iouP[512];
    const int t = threadIdx.x;
    float r = 0.f, io = 0.f;
    if (t < NSB * NLANE) {                                 // 384 items
        int sb = t >> 2, l = t & 3;
        int b  = sb % NBATCH;
        int row = rowsWS[branch * NSB * NLANE + sb * NLANE + l];
        const float* pm = preds + ((size_t)sb * NPRI + row) * DD;
        const float* tg = gt + ((size_t)b * NLANE + l) * DD;
        const float sc[4] = {71.f, 799.f, 180.f, 71.f};
        float sm = 0.f;
        #pragma unroll
        for (int j = 0; j < 4; ++j) {
            float d  = (pm[2 + j] - tg[2 + j]) * sc[j];
            float ad = fabsf(d);
            sm += (ad < 1.f) ? 0.5f * d * d : ad - 0.5f;
        }
        r = sm * 0.25f * 0.25f * (1.0f / 96.0f);           // mean4, /L, /(B*S)
        float so = 0.f, su = 0.f;
        for (int k = 0; k < NPTS; ++k) {
            float rp = pm[6 + k] * 799.f;
            float rt = tg[6 + k];
            bool inv = (rt < 0.f) | (rt >= 800.f);
            float ov = fminf(rp + 15.f, rt + 15.f) - fmaxf(rp - 15.f, rt - 15.f);
            float un = fmaxf(rp + 15.f, rt + 15.f) - fminf(rp - 15.f, rt - 15.f);
            so += inv ? 0.f : ov;
            su += inv ? 0.f : un;
        }
        float iou = so / (su + 1e-9f);
        io = (1.0f - iou) * 0.25f * (1.0f / 96.0f);
    }
    regP[t] = r; iouP[t] = io;
    __syncthreads();
    for (int o = 256; o >= 4; o >>= 1) {                   // keeps l-phase separate
        if (t < o) { regP[t] += regP[t + o]; iouP[t] += iouP[t + o]; }
        __syncthreads();
    }
    if (t < NLANE) {                                       // rows_last = (s=2,b=31)=sb 95
        int rl = rowsWS[branch * NSB * NLANE + 95 * NLANE + t];
        lossWS[branch * NPRI + rl] += 0.5f * regP[t] + 2.0f * iouP[t];
    }
}

// ---------------------------------------------------------------------------
// Kernel 4: delta = median(lossA - lossB) via LDS bitonic sort (2048 padded)
// ---------------------------------------------------------------------------
__global__ __launch_bounds__(1024) void median_kernel(
    const float* __restrict__ lossWS, float* __restrict__ deltaWS)
{
    __shared__ float a[2048];
    const int t = threadIdx.x;
    for (int i = t; i < 2048; i += 1024)
        a[i] = (i < NPRI) ? lossWS[i] - lossWS[NPRI + i] : 3.0e38f;
    __syncthreads();
    for (int k = 2; k <= 2048; k <<= 1)
        for (int j = k >> 1; j > 0; j >>= 1) {
            for (int i = t; i < 2048; i += 1024) {
                int ix = i ^ j;
                if (ix > i) {
                    bool up = (i & k) == 0;
                    float x = a[i], y = a[ix];
                    if ((x > y) == up) { a[i] = y; a[ix] = x; }
                }
            }
            __syncthreads();
        }
    if (t == 0) deltaWS[0] = 0.5f * (a[999] + a[1000]);    // even-N median
}

// ---------------------------------------------------------------------------
// Kernel 5: total = sum_n (1-dm)*A' + dm*B'  -- dot product via f32 WMMA
// (A = 16x4 chunk of products, B = ones; diagonal-free: every column = rowsum)
// ---------------------------------------------------------------------------
__global__ __launch_bounds__(32) void final_kernel(
    const float* __restrict__ diff, const float* __restrict__ lossWS,
    const float* __restrict__ deltaWS, float* __restrict__ out)
{
    const int lane = threadIdx.x;
    const float delta = deltaWS[0];
    const float* A_ = lossWS;
    const float* B_ = lossWS + NPRI;
    auto term = [&](int n) -> float {
        if (n >= NPRI) return 0.f;
        float dm = (diff[n] + diff[NPRI + n] + diff[2 * NPRI + n]) * (1.0f / 3.0f);
        return (1.0f - dm) * (A_[n] - 0.5f * delta) + dm * (B_[n] + 0.5f * delta);
    };
#if __has_builtin(__builtin_amdgcn_wmma_f32_16x16x4_f32)
    v8f c = {0.f, 0.f, 0.f, 0.f, 0.f, 0.f, 0.f, 0.f};
    v2f bones; bones.x = 1.0f; bones.y = 1.0f;
    const int m  = lane & 15;            // A row
    const int kb = (lane >> 4) * 2;      // lanes 0-15: K=0,1 ; lanes 16-31: K=2,3
    for (int it = 0; it < 32; ++it) {    // 32*64 = 2048 >= 2000 (zero padded)
        int n0 = it * 64 + m * 4 + kb;
        v2f a; a.x = term(n0); a.y = term(n0 + 1);
        c = __builtin_amdgcn_wmma_f32_16x16x4_f32(
                false, a, false, bones, (short)0, c, false, false);
    }
    float s = c[0] + c[1] + c[2] + c[3] + c[4] + c[5] + c[6] + c[7];
    s += __shfl_xor(s, 16);              // rows 0-7 + rows 8-15
    if (lane == 0) out[0] = s;
#else
    float s = 0.f;
    for (int n = lane; n < NPRI; n += 32) s += term(n);
    #pragma unroll
    for (int m2 = 16; m2 > 0; m2 >>= 1) s += __shfl_xor(s, m2);
    if (lane == 0) out[0] = s;
#endif
}

// ---------------------------------------------------------------------------
extern "C" void kernel_launch(void* const* d_in, const int* in_sizes, int n_in,
                              void* d_out, int out_size, void* d_ws, size_t ws_size,
                              hipStream_t stream) {
    const float* predF = (const float*)d_in[0];
    const float* predS = (const float*)d_in[1];
    const float* gt    = (const float*)d_in[2];
    const float* diff  = (const float*)d_in[3];

    int*   rowsWS  = (int*)d_ws;                         // [2][96][4]  (768 ints)
    float* lossWS  = (float*)d_ws + 2 * NSB * NLANE;     // [2][2000]
    float* deltaWS = lossWS + 2 * NPRI;                  // [1]
    float* out     = (float*)d_out;

    hipLaunchKernelGGL(assign_kernel, dim3(2 * NSB), dim3(256), 0, stream,
                       predF, predS, gt, rowsWS);
    hipLaunchKernelGGL(cls_kernel, dim3(16), dim3(256), 0, stream,
                       predF, predS, rowsWS, lossWS);
    hipLaunchKernelGGL(regiou_kernel, dim3(2), dim3(512), 0, stream,
                       predF, predS, gt, rowsWS, lossWS);
    hipLaunchKernelGGL(median_kernel, dim3(1), dim3(1024), 0, stream,
                       lossWS, deltaWS);
    hipLaunchKernelGGL(final_kernel, dim3(1), dim3(32), 0, stream,
                       diff, lossWS, deltaWS, out);
}